// CapsuleFC_4595615007177
// MI455X (gfx1250) — compile-verified
//
#include <hip/hip_runtime.h>
#include <hip/hip_bf16.h>

#define B_SZ 64
#define N_SZ 4096
#define M_SZ 128
#define CHUNK 64
#define NCHUNK (N_SZ / CHUNK)   // 64

typedef __attribute__((ext_vector_type(2))) float v2f;
typedef __attribute__((ext_vector_type(8))) float v8f;

// ---------------------------------------------------------------------------
// Kernel 1: routing logits qk[b,n,m] = SCALE * sum_{a,x,d} pose*w*ncv
// One wave handles (16 b's) x (16 m's) x CHUNK n's.
// WMMA tile: D[b,m] = sum_x pose[b,n,a,x] * w[n,x,d,m]  (per (a,d))
// then qk_tile += D (*) ncv_tile  elementwise (lane=m, VGPR-row=b).
// ---------------------------------------------------------------------------
__global__ __launch_bounds__(32) void capsfc_qk_kernel(
    const float* __restrict__ x, const float* __restrict__ ncv,
    const float* __restrict__ w, float* __restrict__ qk) {
  const int lane = threadIdx.x;
  const int hf = lane >> 4;        // which K-half this lane holds for A/B
  const int lm = lane & 15;        // row (A: b-row) / col (B,D: m-col)
  const int n0 = blockIdx.x * CHUNK;
  const int m0 = blockIdx.y * 16;
  const int b0 = blockIdx.z * 16;

  // Preload ncv in D-tile layout: cv[a*4+d][r] = ncv[b0+r+8*hf, m0+lm, a, d]
  v8f cv[16];
#pragma unroll
  for (int a = 0; a < 4; ++a)
#pragma unroll
    for (int d = 0; d < 4; ++d)
#pragma unroll
      for (int r = 0; r < 8; ++r)
        cv[a * 4 + d][r] =
            ncv[((b0 + r + 8 * hf) * M_SZ + m0 + lm) * 16 + a * 4 + d];

  for (int n = n0; n < n0 + CHUNK; ++n) {
    // A (16x4 f32): lanes0-15 K=0,1 ; lanes16-31 K=2,3 ; row M = b0+lm
    v2f A[4];
#pragma unroll
    for (int a = 0; a < 4; ++a)
      A[a] = *(const v2f*)&x[((b0 + lm) * N_SZ + n) * 16 + a * 4 + 2 * hf];
    // B (4x16 f32): VGPR0 rows K=0|2, VGPR1 rows K=1|3 ; col N = m0+lm
    v2f Bm[4];
#pragma unroll
    for (int d = 0; d < 4; ++d) {
      const float* wp = &w[((n * 4 + 2 * hf) * 4 + d) * M_SZ + m0 + lm];
      Bm[d][0] = wp[0];
      Bm[d][1] = wp[512];  // next x: stride 4*M_SZ floats
    }
    if (n + 1 < n0 + CHUNK) {
      __builtin_prefetch(&w[((n + 1) * 16) * M_SZ + m0 + lm], 0, 3);
      __builtin_prefetch(&x[((b0 + lm) * N_SZ + n + 1) * 16], 0, 3);
    }
    float qa[8] = {0.f, 0.f, 0.f, 0.f, 0.f, 0.f, 0.f, 0.f};
#pragma unroll
    for (int a = 0; a < 4; ++a) {
#pragma unroll
      for (int d = 0; d < 4; ++d) {
        v8f c = {0.f, 0.f, 0.f, 0.f, 0.f, 0.f, 0.f, 0.f};
        v8f D = __builtin_amdgcn_wmma_f32_16x16x4_f32(
            false, A[a], false, Bm[d], (short)0, c, false, false);
#pragma unroll
        for (int r = 0; r < 8; ++r) qa[r] += D[r] * cv[a * 4 + d][r];
      }
    }
#pragma unroll
    for (int r = 0; r < 8; ++r)
      qk[((b0 + r + 8 * hf) * N_SZ + n) * M_SZ + m0 + lm] = qa[r] * 0.25f;
  }
}

// ---------------------------------------------------------------------------
// Kernel 2: per-(b,n) softmax over m=128, then renormalize by (sum + 1e-10),
// in place. One wave32 per row; 4 values per lane; __shfl_xor reductions.
// ---------------------------------------------------------------------------
__global__ __launch_bounds__(256) void capsfc_softmax_kernel(
    float* __restrict__ qk) {
  const long row = (long)((blockIdx.x * 256 + threadIdx.x) >> 5);
  const int lane = threadIdx.x & 31;
  float* rp = qk + row * M_SZ + lane * 4;
  float v0 = rp[0], v1 = rp[1], v2 = rp[2], v3 = rp[3];
  float mx = fmaxf(fmaxf(v0, v1), fmaxf(v2, v3));
#pragma unroll
  for (int off = 16; off > 0; off >>= 1)
    mx = fmaxf(mx, __shfl_xor(mx, off, 32));
  v0 = __expf(v0 - mx);
  v1 = __expf(v1 - mx);
  v2 = __expf(v2 - mx);
  v3 = __expf(v3 - mx);
  float s = v0 + v1 + v2 + v3;
#pragma unroll
  for (int off = 16; off > 0; off >>= 1) s += __shfl_xor(s, off, 32);
  float inv = 1.f / s;
  v0 *= inv; v1 *= inv; v2 *= inv; v3 *= inv;
  float s2 = v0 + v1 + v2 + v3;
#pragma unroll
  for (int off = 16; off > 0; off >>= 1) s2 += __shfl_xor(s2, off, 32);
  float inv2 = 1.f / (s2 + 1e-10f);
  rp[0] = v0 * inv2;
  rp[1] = v1 * inv2;
  rp[2] = v2 * inv2;
  rp[3] = v3 * inv2;
}

// ---------------------------------------------------------------------------
// Kernel 3: out_partial[b,m,a,d] += qk'[b,n,m] * V[b,n,a,d,m] over an n-chunk.
// Votes recomputed with the same WMMA tiling; accumulators live in the exact
// D-tile layout (16 tiles x v8f = 128 VGPRs).
// ---------------------------------------------------------------------------
__global__ __launch_bounds__(32) void capsfc_agg_kernel(
    const float* __restrict__ x, const float* __restrict__ w,
    const float* __restrict__ qk, float* __restrict__ partial) {
  const int lane = threadIdx.x;
  const int hf = lane >> 4;
  const int lm = lane & 15;
  const int n0 = blockIdx.x * CHUNK;
  const int m0 = blockIdx.y * 16;
  const int b0 = blockIdx.z * 16;

  v8f acc[16];
#pragma unroll
  for (int t = 0; t < 16; ++t)
#pragma unroll
    for (int r = 0; r < 8; ++r) acc[t][r] = 0.f;

  for (int n = n0; n < n0 + CHUNK; ++n) {
    v2f A[4];
#pragma unroll
    for (int a = 0; a < 4; ++a)
      A[a] = *(const v2f*)&x[((b0 + lm) * N_SZ + n) * 16 + a * 4 + 2 * hf];
    v2f Bm[4];
#pragma unroll
    for (int d = 0; d < 4; ++d) {
      const float* wp = &w[((n * 4 + 2 * hf) * 4 + d) * M_SZ + m0 + lm];
      Bm[d][0] = wp[0];
      Bm[d][1] = wp[512];
    }
    float qw[8];
#pragma unroll
    for (int r = 0; r < 8; ++r)
      qw[r] = qk[((b0 + r + 8 * hf) * N_SZ + n) * M_SZ + m0 + lm];
#pragma unroll
    for (int a = 0; a < 4; ++a) {
#pragma unroll
      for (int d = 0; d < 4; ++d) {
        v8f c = {0.f, 0.f, 0.f, 0.f, 0.f, 0.f, 0.f, 0.f};
        v8f D = __builtin_amdgcn_wmma_f32_16x16x4_f32(
            false, A[a], false, Bm[d], (short)0, c, false, false);
#pragma unroll
        for (int r = 0; r < 8; ++r) acc[a * 4 + d][r] += D[r] * qw[r];
      }
    }
  }
  const long base =
      (long)((blockIdx.z * 8 + blockIdx.y) * NCHUNK + blockIdx.x) * 4096;
#pragma unroll
  for (int t = 0; t < 16; ++t)
#pragma unroll
    for (int r = 0; r < 8; ++r)
      partial[base + (t * 8 + r) * 32 + lane] = acc[t][r];
}

// ---------------------------------------------------------------------------
// Kernel 4: reduce n-chunk partials and LayerNorm over the 16 output dims.
// One thread per (b, m).
// ---------------------------------------------------------------------------
__global__ __launch_bounds__(256) void capsfc_reduce_ln_kernel(
    const float* __restrict__ partial, const float* __restrict__ gamma,
    const float* __restrict__ beta, float* __restrict__ out) {
  const int tid = blockIdx.x * 256 + threadIdx.x;  // 0..8191
  const int b = tid >> 7, m = tid & 127;
  const int bg = b >> 4, bi = b & 15, r = bi & 7, hf = bi >> 3;
  const int mg = m >> 4, lm = m & 15, ln_ = hf * 16 + lm;

  float vals[16];
  float mean = 0.f;
#pragma unroll
  for (int t = 0; t < 16; ++t) {
    float s = 0.f;
    const float* p =
        partial + (long)(bg * 8 + mg) * NCHUNK * 4096 + (t * 8 + r) * 32 + ln_;
    for (int c = 0; c < NCHUNK; ++c) s += p[(long)c * 4096];
    vals[t] = s;
    mean += s;
  }
  mean *= (1.f / 16.f);
  float var = 0.f;
#pragma unroll
  for (int t = 0; t < 16; ++t) {
    float d = vals[t] - mean;
    var += d * d;
  }
  var *= (1.f / 16.f);
  const float rstd = rsqrtf(var + 1e-5f);
#pragma unroll
  for (int t = 0; t < 16; ++t)
    out[(long)tid * 16 + t] = (vals[t] - mean) * rstd * gamma[t] + beta[t];
}

// ---------------------------------------------------------------------------
extern "C" void kernel_launch(void* const* d_in, const int* in_sizes, int n_in,
                              void* d_out, int out_size, void* d_ws,
                              size_t ws_size, hipStream_t stream) {
  const float* x     = (const float*)d_in[0];  // (B,N,16)
  const float* ncv   = (const float*)d_in[1];  // (B,M,16)
  const float* w     = (const float*)d_in[2];  // (N,4,4,M)
  const float* gamma = (const float*)d_in[3];  // (16)
  const float* beta  = (const float*)d_in[4];  // (16)
  float* out = (float*)d_out;                  // (B,M,16)

  float* qk = (float*)d_ws;                               // B*N*M f32 = 128 MiB
  float* partial = qk + (size_t)B_SZ * N_SZ * M_SZ;       // 2048*4096 f32 = 32 MiB

  dim3 grid_wmma(NCHUNK, M_SZ / 16, B_SZ / 16);  // (64, 8, 4) -> 2048 waves
  capsfc_qk_kernel<<<grid_wmma, 32, 0, stream>>>(x, ncv, w, qk);

  const int rows = B_SZ * N_SZ;  // 262144 rows, 1 wave each
  capsfc_softmax_kernel<<<(rows * 32) / 256, 256, 0, stream>>>(qk);

  capsfc_agg_kernel<<<grid_wmma, 32, 0, stream>>>(x, w, qk, partial);

  capsfc_reduce_ln_kernel<<<(B_SZ * M_SZ) / 256, 256, 0, stream>>>(
      partial, gamma, beta, out);
}